// DN_21758304321887
// MI455X (gfx1250) — compile-verified
//
#include <hip/hip_runtime.h>
#include <hip/hip_bf16.h>
#include <math.h>

// ---------------- problem constants ----------------
#define Yc 4096
#define Xc 10000
#define Zc 100

constexpr size_t O_WX2Y = 0ULL;
constexpr size_t O_YNUM = 40960000ULL;
constexpr size_t O_WZ2Y = 40960001ULL;
constexpr size_t O_WY2Z = 41369601ULL;
constexpr size_t O_DIFF = 41779201ULL;
constexpr size_t O_FAC  = 82739201ULL;
constexpr size_t O_YAGE = 123699201ULL;
constexpr size_t O_ZAGE = 123703297ULL;
constexpr size_t O_THR  = 123703397ULL;

typedef __attribute__((ext_vector_type(2))) float v2f;
typedef __attribute__((ext_vector_type(8))) float v8f;

// ---------------- K1: x_n = x / max(||x||, 1e-12) ----------------
__global__ void k_xnorm(const float* __restrict__ x, float* __restrict__ xn) {
    __shared__ float red[256];
    int t = threadIdx.x;
    float s = 0.f;
    for (int k = t; k < Xc; k += 256) { float v = x[k]; s += v * v; }
    red[t] = s; __syncthreads();
    for (int o = 128; o > 0; o >>= 1) { if (t < o) red[t] += red[t + o]; __syncthreads(); }
    float inv = 1.0f / fmaxf(sqrtf(red[0]), 1e-12f);
    for (int k = t; k < Xc; k += 256) xn[k] = x[k] * inv;
}

// ---------------- K2: y_bu via V_WMMA_F32_16X16X4_F32 ----------------
// Block = 256 threads (8 waves) owns 16 rows. Per 64-wide K chunk:
//  - cooperative float2 (b64) load of wf = w*factor into LDS (stride-66 pad);
//    thread t owns col-pair c0 of rows (t>>5) and (t>>5)+8, accumulating
//    sum(wf^2) per row in two VALU registers; 156 guard-free full chunks
//    plus one guarded 16-column tail chunk,
//  - each wave runs exactly two straight-line WMMA k-steps (steps wid and
//    wid+8), keeping control flow uniform so EXEC is provably all-ones:
//    A = x-chunk replicated over the 16 A rows, B = LDS wf tile,
//    so D[i][n] = dot(x_chunk, wf_row_n) for every i.
#define CH 64
#define STR 66
__global__ void k_ybu(const float* __restrict__ w, const float* __restrict__ f,
                      const float* __restrict__ xn, float* __restrict__ ybu) {
    __shared__ __align__(16) float wfs[16 * STR];
    __shared__ __align__(16) float xs[CH];
    __shared__ float dot_s[16];
    __shared__ float s1_s[16];
    const int t = threadIdx.x;
    const int lane = t & 31;
    const int wid = t >> 5;
    const int j0 = blockIdx.x * 16;

    if (t < 16) { dot_s[t] = 0.f; s1_s[t] = 0.f; }
    float s1a0 = 0.f, s1a1 = 0.f;
    v8f cacc = {0.f, 0.f, 0.f, 0.f, 0.f, 0.f, 0.f, 0.f};

    const int hi2 = (lane >> 4) << 1;       // 0 for lanes 0-15, 2 for lanes 16-31
    const int rr  = (lane & 15);
    const int kkA = wid * 4 + hi2;          // this wave's first k-step
    const int kkB = kkA + 32;               // this wave's second k-step
    const int r0  = t >> 5;                 // tile rows r0 and r0+8
    const int c0  = (t & 31) * 2;           // tile column pair

    __syncthreads();
    const int NFULL = Xc / CH;              // 156 guard-free chunks
    for (int ch = 0; ch < NFULL; ++ch) {
        const int kb = ch * CH;
        if (t < 32) *(float2*)&xs[t * 2] = *(const float2*)&xn[kb + t * 2];
        #pragma unroll
        for (int i = 0; i < 2; ++i) {
            int r = r0 + 8 * i;
            size_t off = (size_t)(j0 + r) * Xc + (size_t)(kb + c0);
            float2 wv = *(const float2*)&w[off];
            float2 fv = *(const float2*)&f[off];
            float a = wv.x * fv.x, b = wv.y * fv.y;
            float2 ab; ab.x = a; ab.y = b;
            *(float2*)&wfs[r * STR + c0] = ab;
            float sq = a * a + b * b;
            if (i == 0) s1a0 += sq; else s1a1 += sq;
        }
        __syncthreads();
        // two K=4 WMMA steps per wave, straight-line (uniform EXEC)
        {
            v2f a0 = *(const v2f*)&xs[kkA];
            v2f b0 = *(const v2f*)&wfs[rr * STR + kkA];
            cacc = __builtin_amdgcn_wmma_f32_16x16x4_f32(
                false, a0, false, b0, (short)0, cacc, false, false);
            v2f a1 = *(const v2f*)&xs[kkB];
            v2f b1 = *(const v2f*)&wfs[rr * STR + kkB];
            cacc = __builtin_amdgcn_wmma_f32_16x16x4_f32(
                false, a1, false, b1, (short)0, cacc, false, false);
        }
        __syncthreads();
    }
    // tail chunk: 16 valid columns (float2-clean: a pair is all-in or all-out)
    {
        const int kb = NFULL * CH;
        const int rem = Xc - kb;            // 16
        if (t < 32) {
            float2 xv; xv.x = 0.f; xv.y = 0.f;
            if (t * 2 < rem) xv = *(const float2*)&xn[kb + t * 2];
            *(float2*)&xs[t * 2] = xv;
        }
        #pragma unroll
        for (int i = 0; i < 2; ++i) {
            int r = r0 + 8 * i;
            float2 ab; ab.x = 0.f; ab.y = 0.f;
            if (c0 < rem) {
                size_t off = (size_t)(j0 + r) * Xc + (size_t)(kb + c0);
                float2 wv = *(const float2*)&w[off];
                float2 fv = *(const float2*)&f[off];
                ab.x = wv.x * fv.x; ab.y = wv.y * fv.y;
            }
            *(float2*)&wfs[r * STR + c0] = ab;
            float sq = ab.x * ab.x + ab.y * ab.y;
            if (i == 0) s1a0 += sq; else s1a1 += sq;
        }
        __syncthreads();
        {
            v2f a0 = *(const v2f*)&xs[kkA];
            v2f b0 = *(const v2f*)&wfs[rr * STR + kkA];
            cacc = __builtin_amdgcn_wmma_f32_16x16x4_f32(
                false, a0, false, b0, (short)0, cacc, false, false);
            v2f a1 = *(const v2f*)&xs[kkB];
            v2f b1 = *(const v2f*)&wfs[rr * STR + kkB];
            cacc = __builtin_amdgcn_wmma_f32_16x16x4_f32(
                false, a1, false, b1, (short)0, cacc, false, false);
        }
        __syncthreads();
    }
    // merge wave partials: every D row is identical; row M=0 lives in cacc[0]
    if (lane < 16) atomicAdd(&dot_s[lane], cacc[0]);
    atomicAdd(&s1_s[r0],     s1a0);
    atomicAdd(&s1_s[r0 + 8], s1a1);
    __syncthreads();
    if (t < 16) ybu[j0 + t] = dot_s[t] / fmaxf(sqrtf(s1_s[t]), 1e-12f);
}

// ---------------- K2b: normalize w_z2y rows, write to out, compute pre ----------------
__global__ void k_z2y(const float* __restrict__ wz2y, const float* __restrict__ ybu,
                      const int* __restrict__ z, float* __restrict__ out,
                      float* __restrict__ pre) {
    __shared__ float red[128];
    int t = threadIdx.x, j = blockIdx.x;
    float v = (t < Zc) ? wz2y[j * Zc + t] : 0.f;
    red[t] = v * v; __syncthreads();
    for (int o = 64; o > 0; o >>= 1) { if (t < o) red[t] += red[t + o]; __syncthreads(); }
    float inv = 1.0f / fmaxf(sqrtf(red[0]), 1e-12f);
    if (t < Zc) out[O_WZ2Y + (size_t)j * Zc + t] = v * inv;
    if (t == 0) {
        int c = z[0];
        pre[j] = 0.5f * ybu[j] + 0.5f * (wz2y[j * Zc + c] * inv);
    }
}

// ---------------- K3: top-k competition + small updates ----------------
__global__ void k_compete(const float* __restrict__ pre, const int* __restrict__ y_age,
                          const float* __restrict__ y_thresh, const int* __restrict__ z,
                          const int* __restrict__ z_age, float* __restrict__ out,
                          int* __restrict__ s_iw, float* __restrict__ s_fw) {
    __shared__ float bv[256]; __shared__ int bi[256];
    __shared__ float av[256]; __shared__ int ai[256];
    __shared__ int cnt[256];
    __shared__ int s_idx; __shared__ float s_lr; __shared__ float s_mr;
    int t = threadIdx.x;
    float lbv = -3.4e38f; int lbi = 0;
    float lav = -3.4e38f; int lai = 0;
    int lc = 0;
    for (int j = t; j < Yc; j += 256) {
        float p = pre[j];
        if (p > lbv) { lbv = p; lbi = j; }
        int un = (y_age[j] < 1);
        lc += un;
        float pu = p * (un ? 1.0f : 0.0f);
        if (pu > lav) { lav = pu; lai = j; }
    }
    bv[t] = lbv; bi[t] = lbi; av[t] = lav; ai[t] = lai; cnt[t] = lc;
    __syncthreads();
    for (int o = 128; o > 0; o >>= 1) {
        if (t < o) {
            if (bv[t + o] > bv[t] || (bv[t + o] == bv[t] && bi[t + o] < bi[t])) { bv[t] = bv[t + o]; bi[t] = bi[t + o]; }
            if (av[t + o] > av[t] || (av[t + o] == av[t] && ai[t + o] < ai[t])) { av[t] = av[t + o]; ai[t] = ai[t + o]; }
            cnt[t] += cnt[t + o];
        }
        __syncthreads();
    }
    if (t == 0) {
        int idx0 = bi[0]; float mr = bv[0];
        bool keep = (mr > y_thresh[idx0]) || ((float)y_age[idx0] < 1.0f);
        bool hasun = cnt[0] > 0;
        int idx = (keep || !hasun) ? idx0 : ai[0];
        float lr = 1.0f / ((float)y_age[idx] + 1.0f);
        s_idx = idx; s_lr = lr; s_mr = mr;
        s_iw[0] = idx; s_fw[0] = lr;
    }
    __syncthreads();
    int idx = s_idx; float lr = s_lr; float mr = s_mr;
    int lact = 0;
    for (int j = t; j < Yc; j += 256) {
        float th = y_thresh[j];
        if (j == idx) th = lr * mr + (1.0f - lr) * th;
        out[O_THR + j] = th;
        float ag = (float)y_age[j] + ((j == idx) ? 1.0f : 0.0f);
        out[O_YAGE + j] = ag;
        lact += (ag >= 1.0f) ? 1 : 0;
    }
    cnt[t] = lact; __syncthreads();
    for (int o = 128; o > 0; o >>= 1) { if (t < o) cnt[t] += cnt[t + o]; __syncthreads(); }
    if (t == 0) out[O_YNUM] = (float)cnt[0];
    int c = z[0];
    if (t < Zc) out[O_ZAGE + t] = (float)z_age[t] + ((t == c) ? 1.0f : 0.0f);
    if (t < Zc) {
        size_t p = O_WZ2Y + (size_t)idx * Zc + t;       // already normalized by K2b
        float vn = out[p];
        out[p] = (1.0f - lr) * vn + lr * ((t == c) ? 1.0f : 0.0f);
    }
}

// ---------------- K4: stream w_x2y / diff / factor (LDS-staged diff row) ----------------
__global__ void k_stream(const float* __restrict__ w, const float* __restrict__ diff,
                         const float* __restrict__ fac, const int* __restrict__ y_age,
                         const float* __restrict__ xn, const int* __restrict__ s_iw,
                         const float* __restrict__ s_fw, float* __restrict__ out) {
    __shared__ __align__(16) float sd[Xc];
    __shared__ float red[256];
    const int t = threadIdx.x;
    const int j = blockIdx.x;
    const int idx = s_iw[0];
    const float lr = s_fw[0];
    const bool isidx = (j == idx);
    const size_t base = (size_t)j * Xc;

    const float4* w4 = (const float4*)(w + base);
    const float4* d4 = (const float4*)(diff + base);
    const float4* x4 = (const float4*)xn;
    float4* ow4 = (float4*)(out + O_WX2Y + base);       // offset 0: 16B aligned
    float*  od  = out + O_DIFF + base;                  // odd offset: scalar stores
    float4* sd4 = (float4*)sd;

    float lsum = 0.f;
    for (int q = t; q < Xc / 4; q += 256) {
        if (q + 256 < Xc / 4) {                         // gfx1250 global_prefetch_b8
            __builtin_prefetch(&w4[q + 256], 0, 0);
            __builtin_prefetch(&d4[q + 256], 0, 0);
        }
        float4 wv = w4[q];
        float4 dv = d4[q];
        if (isidx) {
            float4 xv = x4[q];
            wv.x = (1.f - lr) * wv.x + lr * xv.x;
            wv.y = (1.f - lr) * wv.y + lr * xv.y;
            wv.z = (1.f - lr) * wv.z + lr * xv.z;
            wv.w = (1.f - lr) * wv.w + lr * xv.w;
            dv.x = (1.f - lr) * dv.x + lr * fabsf(wv.x - dv.x);
            dv.y = (1.f - lr) * dv.y + lr * fabsf(wv.y - dv.y);
            dv.z = (1.f - lr) * dv.z + lr * fabsf(wv.z - dv.z);
            dv.w = (1.f - lr) * dv.w + lr * fabsf(wv.w - dv.w);
        }
        ow4[q] = wv;
        int k = q * 4;
        od[k] = dv.x; od[k + 1] = dv.y; od[k + 2] = dv.z; od[k + 3] = dv.w;
        sd4[q] = dv;
        lsum += dv.x + dv.y + dv.z + dv.w;
    }

    if ((float)y_age[j] > 20.0f) {                      // synapse maintenance row
        red[t] = lsum; __syncthreads();
        for (int o = 128; o > 0; o >>= 1) { if (t < o) red[t] += red[t + o]; __syncthreads(); }
        float mean  = red[0] / (float)Xc;
        float lower = 0.8f * mean, upper = 1.2f * mean;
        float denom = fmaxf(lower - upper, 1e-12f);
        float* ofa = out + O_FAC + base;
        for (int k = t; k < Xc; k += 256) {
            float dn  = sd[k];
            float uf  = (dn > upper) ? 0.f : 1.f;
            float mid = (dn - upper) / denom;
            ofa[k] = (dn < lower) ? 1.f : mid * uf;
        }
    } else {                                            // factor row passes through
        const float4* f4 = (const float4*)(fac + base);
        float* ofa = out + O_FAC + base;
        for (int q = t; q < Xc / 4; q += 256) {
            float4 fv = f4[q];
            int k = q * 4;
            ofa[k] = fv.x; ofa[k + 1] = fv.y; ofa[k + 2] = fv.z; ofa[k + 3] = fv.w;
        }
    }
}

// ---------------- K5: normalize w_y2z rows + row c update ----------------
__global__ void k_y2z(const float* __restrict__ wy2z, const int* __restrict__ z,
                      const int* __restrict__ z_age, const int* __restrict__ s_iw,
                      float* __restrict__ out) {
    __shared__ float red[256];
    int t = threadIdx.x, r = blockIdx.x;
    size_t base = (size_t)r * Yc;
    float s = 0.f;
    for (int k = t; k < Yc; k += 256) { float v = wy2z[base + k]; s += v * v; }
    red[t] = s; __syncthreads();
    for (int o = 128; o > 0; o >>= 1) { if (t < o) red[t] += red[t + o]; __syncthreads(); }
    float inv = 1.0f / fmaxf(sqrtf(red[0]), 1e-12f);
    int c = z[0];
    if (r == c) {
        int idx = s_iw[0];
        float lrz = 1.0f / ((float)z_age[c] + 1.0f);
        for (int k = t; k < Yc; k += 256) {
            float vn = wy2z[base + k] * inv;
            out[O_WY2Z + base + k] = (1.0f - lrz) * vn + lrz * ((k == idx) ? 1.0f : 0.0f);
        }
    } else {
        for (int k = t; k < Yc; k += 256)
            out[O_WY2Z + base + k] = wy2z[base + k] * inv;
    }
}

// ---------------- launch ----------------
extern "C" void kernel_launch(void* const* d_in, const int* in_sizes, int n_in,
                              void* d_out, int out_size, void* d_ws, size_t ws_size,
                              hipStream_t stream) {
    const float* x      = (const float*)d_in[0];
    const int*   z      = (const int*)  d_in[1];
    const float* w_x2y  = (const float*)d_in[2];
    const float* w_z2y  = (const float*)d_in[3];
    const float* w_y2z  = (const float*)d_in[4];
    const float* diff   = (const float*)d_in[5];
    const float* fac    = (const float*)d_in[6];
    const int*   y_age  = (const int*)  d_in[7];
    const float* y_thr  = (const float*)d_in[8];
    const int*   z_age  = (const int*)  d_in[9];
    float* out = (float*)d_out;

    float* wsf = (float*)d_ws;
    float* xn  = wsf;                 // [10000]
    float* ybu = wsf + 10000;         // [4096]
    float* pre = wsf + 14096;         // [4096]
    int*   s_iw = (int*)(wsf + 18192);  // idx
    float* s_fw = wsf + 18193;          // lr

    k_xnorm  <<<1,        256, 0, stream>>>(x, xn);
    k_ybu    <<<Yc / 16,  256, 0, stream>>>(w_x2y, fac, xn, ybu);
    k_z2y    <<<Yc,       128, 0, stream>>>(w_z2y, ybu, z, out, pre);
    k_compete<<<1,        256, 0, stream>>>(pre, y_age, y_thr, z, z_age, out, s_iw, s_fw);
    k_stream <<<Yc,       256, 0, stream>>>(w_x2y, diff, fac, y_age, xn, s_iw, s_fw, out);
    k_y2z    <<<Zc,       256, 0, stream>>>(w_y2z, z, z_age, s_iw, out);
}